// SobelPatchScorer_41326175322551
// MI455X (gfx1250) — compile-verified
//
#include <hip/hip_runtime.h>
#include <stdint.h>

#define IMG_W 448
#define IMG_H 448
#define HW (448 * 448)
#define TILE 64
#define GRAY_W 66     // 64 + 2 halo
#define GSTRIDE 68    // padded LDS row stride (floats)

#if __has_builtin(__builtin_amdgcn_global_load_async_to_lds_b32)
#define HAVE_ASYNC 1
#else
#define HAVE_ASYNC 0
#endif

typedef __attribute__((address_space(1))) int* global_i32_ptr;
typedef __attribute__((address_space(3))) int* lds_i32_ptr;

// Async DMA one f32 from global to LDS (gfx1250 GLOBAL_LOAD_ASYNC_TO_LDS_B32,
// tracked by ASYNCcnt). Pointers go through uintptr_t to form addrspace(1)/(3)
// pointers (low 32 bits of a generic LDS pointer are the LDS byte offset).
__device__ __forceinline__ void async_copy_f32(const float* gsrc, float* ldst) {
#if HAVE_ASYNC
  __builtin_amdgcn_global_load_async_to_lds_b32(
      (global_i32_ptr)(uintptr_t)gsrc,
      (lds_i32_ptr)(uint32_t)(uintptr_t)ldst,
      /*imm offset*/ 0, /*cpol*/ 0);
#else
  *ldst = *gsrc;
#endif
}

__device__ __forceinline__ void async_wait_all() {
#if HAVE_ASYNC
#if __has_builtin(__builtin_amdgcn_s_wait_asynccnt)
  __builtin_amdgcn_s_wait_asynccnt(0);
#else
  asm volatile("s_wait_asynccnt 0" ::: "memory");
#endif
#endif
}

// Raw v_sqrt_f32 (~1 ulp) instead of libm sqrtf's v_sqrt_f32 + IEEE fixup
// sequence; precision is far below the 1e-6 epsilon inside the sqrt anyway.
__device__ __forceinline__ float fast_sqrtf(float x) {
#if __has_builtin(__builtin_amdgcn_sqrtf)
  return __builtin_amdgcn_sqrtf(x);
#else
  return sqrtf(x);
#endif
}

__global__ __launch_bounds__(256) void sobel_patch_kernel(
    const float* __restrict__ in, float* __restrict__ out) {
  // 3 channel tiles, 66 rows x 68-float stride each: 3*66*68*4 = 53,856 B
  __shared__ float s_chan[3][GRAY_W * GSTRIDE];

  const int tid = threadIdx.x;
  const int bx = blockIdx.x, by = blockIdx.y, iz = blockIdx.z;
  const int x0 = bx * TILE, y0 = by * TILE;
  const float* img = in + (size_t)iz * (3 * HW);

  // ---- Stage 66x66 halo tile of R,G,B into LDS via async DMA ----
  for (int e = tid; e < GRAY_W * GRAY_W; e += 256) {
    const int ly = e / GRAY_W;
    const int lx = e - ly * GRAY_W;
    const int gx = x0 - 1 + lx;
    const int gy = y0 - 1 + ly;
    const int l = ly * GSTRIDE + lx;
    if ((unsigned)gx < (unsigned)IMG_W && (unsigned)gy < (unsigned)IMG_H) {
      const size_t off = (size_t)gy * IMG_W + gx;
      async_copy_f32(img + off,           &s_chan[0][l]);
      async_copy_f32(img + HW + off,      &s_chan[1][l]);
      async_copy_f32(img + 2 * HW + off,  &s_chan[2][l]);
    } else {
      // SAME zero padding at image borders
      s_chan[0][l] = 0.0f;
      s_chan[1][l] = 0.0f;
      s_chan[2][l] = 0.0f;
    }
  }
  async_wait_all();
  __syncthreads();

  // ---- Grayscale in place into channel 0, vectorized float4 over the ----
  // ---- whole padded plane (contiguous; pad cols computed but unused) ----
  {
    const int NV = (GRAY_W * GSTRIDE) / 4;  // 1122 float4 chunks
    const float4* r4 = (const float4*)s_chan[0];
    const float4* g4 = (const float4*)s_chan[1];
    const float4* b4 = (const float4*)s_chan[2];
    float4* y4 = (float4*)s_chan[0];
    for (int v = tid; v < NV; v += 256) {
      const float4 r = r4[v];
      const float4 g = g4[v];
      const float4 b = b4[v];
      float4 y;
      y.x = 0.299f * r.x + 0.587f * g.x + 0.114f * b.x;
      y.y = 0.299f * r.y + 0.587f * g.y + 0.114f * b.y;
      y.z = 0.299f * r.z + 0.587f * g.z + 0.114f * b.z;
      y.w = 0.299f * r.w + 0.587f * g.w + 0.114f * b.w;
      y4[v] = y;  // element-wise in place: hazard-free
    }
  }
  __syncthreads();

  // ---- Sobel + accumulate, rolling 3x3 register window down 16 rows ----
  const float* gray = s_chan[0];
  const int tx = tid & 63;        // column within tile (0..63)
  const int band = tid >> 6;      // patch row within tile (0..3); constant per wave
  const int R = band * 16;        // first region row this thread sweeps
  const int c = tx;               // left tap column in gray coords

  float a0 = gray[R * GSTRIDE + c];
  float a1 = gray[R * GSTRIDE + c + 1];
  float a2 = gray[R * GSTRIDE + c + 2];
  float b0 = gray[(R + 1) * GSTRIDE + c];
  float b1 = gray[(R + 1) * GSTRIDE + c + 1];
  float b2 = gray[(R + 1) * GSTRIDE + c + 2];

  float acc = 0.0f;
#pragma unroll
  for (int k = 0; k < 16; ++k) {
    const float* row2 = gray + (R + k + 2) * GSTRIDE + c;
    const float c0 = row2[0];
    const float c1 = row2[1];
    const float c2 = row2[2];
    const float gxv = (a2 - a0) + 2.0f * (b2 - b0) + (c2 - c0);
    const float gyv = (c0 - a0) + 2.0f * (c1 - a1) + (c2 - a2);
    acc += fast_sqrtf(gxv * gxv + gyv * gyv + 1e-6f);
    a0 = b0; a1 = b1; a2 = b2;
    b0 = c0; b1 = c1; b2 = c2;
  }

  // ---- Pool: each 16-lane half-wave == exactly one 16x16 patch ----
  acc += __shfl_xor(acc, 1);
  acc += __shfl_xor(acc, 2);
  acc += __shfl_xor(acc, 4);
  acc += __shfl_xor(acc, 8);

  if ((tid & 15) == 0) {
    const int px = tx >> 4;                 // patch col within tile (0..3)
    const int opy = by * 4 + band;          // global patch row (0..27)
    const int opx = bx * 4 + px;            // global patch col (0..27)
    out[(size_t)iz * 784 + opy * 28 + opx] = acc * (1.0f / 256.0f);
  }
}

extern "C" void kernel_launch(void* const* d_in, const int* in_sizes, int n_in,
                              void* d_out, int out_size, void* d_ws, size_t ws_size,
                              hipStream_t stream) {
  const float* imgs = (const float*)d_in[0];
  float* out = (float*)d_out;
  dim3 grid(IMG_W / TILE, IMG_H / TILE, 64);  // (7, 7, 64)
  sobel_patch_kernel<<<grid, dim3(256), 0, stream>>>(imgs, out);
}